// ManifoldEmbedding_64355789964007
// MI455X (gfx1250) — compile-verified
//
#include <hip/hip_runtime.h>
#include <hip/hip_bf16.h>
#include <math.h>

typedef __attribute__((ext_vector_type(2))) float v2f;
typedef __attribute__((ext_vector_type(8))) float v8f;

#define Bn 4
#define Nn 2048
#define Ln 336
#define Hn 8
#define En 5
#define TAUn 4
#define DM 64
#define NME 320          // L - TAU*(E-1)
#define KD 32            // LOW_RANK_D
#define XS (Ln*Hn)       // 2688 floats per sample

// ---------------------------------------------------------------------------
// K1: emb = gelu( conv_proj(x) )  — fused dilated conv + projection
// grid: 8192 blocks (one per (b,n)), block: 320 threads
// ---------------------------------------------------------------------------
__global__ void emb_kernel(const float* __restrict__ x,
                           const float* __restrict__ conv_w,
                           const float* __restrict__ proj_w,
                           const float* __restrict__ proj_b,
                           float* __restrict__ emb) {
    __shared__ float xs[XS];
    __shared__ float pw[NME];
    __shared__ float part[320];
    __shared__ float sv[40];
    const int bn = blockIdx.x;
    const int t  = threadIdx.x;
    const float* xp = x + (size_t)bn * XS;
    for (int i = t; i < XS; i += 320) xs[i] = xp[i];
    pw[t % NME] = proj_w[t % NME];           // t<320 == NME, each slot covered
    __syncthreads();

    // s[h,e] = sum_l pw[l] * x[l+4e, h];  pair p = h*5+e, 8 partial lanes each
    const int p = t >> 3, i0 = t & 7;
    const int h = p / En, e = p % En;
    float acc = 0.f;
    for (int l = i0; l < NME; l += 8)
        acc += pw[l] * xs[(l + TAUn * e) * Hn + h];
    part[t] = acc;
    __syncthreads();
    if (t < 40) {
        float s = 0.f;
        #pragma unroll
        for (int q = 0; q < 8; ++q) s += part[t * 8 + q];
        sv[t] = s;
    }
    __syncthreads();
    if (t < DM) {
        float a = proj_b[0];
        #pragma unroll 8
        for (int q = 0; q < 40; ++q) a += conv_w[t * 40 + q] * sv[q];
        float g = 0.5f * a * (1.0f + erff(a * 0.70710678118654752f)); // exact GELU
        emb[(size_t)bn * DM + t] = g;
    }
}

// ---------------------------------------------------------------------------
// K2: centered projections, normalized up front: Vn = (vx@W0)/||.||, Vhn same
// grid: 8192 blocks, block: 64 threads (wave0 -> W0, wave1 -> U0)
// ---------------------------------------------------------------------------
__global__ void vec_kernel(const float* __restrict__ target,
                           const float* __restrict__ W0,
                           const float* __restrict__ U0,
                           float* __restrict__ Vn,
                           float* __restrict__ Vhn) {
    __shared__ float tl[Ln];
    __shared__ float red[64];
    __shared__ float smu;
    const int bn = blockIdx.x;
    const int t  = threadIdx.x;
    const float* tp = target + (size_t)bn * Ln;
    for (int i = t; i < Ln; i += 64) tl[i] = tp[i];
    __syncthreads();
    float ps = 0.f;
    for (int i = t; i < Ln; i += 64) ps += tl[i];
    red[t] = ps;
    __syncthreads();
    if (t == 0) {
        float s = 0.f;
        for (int q = 0; q < 64; ++q) s += red[q];
        smu = s / (float)Ln;
    }
    __syncthreads();
    const float mu = smu;
    const float* M = (t < 32) ? W0 : U0;     // uniform per wave
    const int d = t & 31;
    float v = 0.f;
    for (int l = 0; l < Ln; ++l) v += (tl[l] - mu) * M[l * KD + d];
    float vv = v * v;
    #pragma unroll
    for (int o = 16; o > 0; o >>= 1) vv += __shfl_xor(vv, o, 32);
    const float inv = 1.0f / sqrtf(vv + 1e-6f);
    float* dst = (t < 32) ? Vn : Vhn;
    dst[(size_t)bn * KD + d] = v * inv;
}

// ---------------------------------------------------------------------------
// K3: pearson_cor[b] = Vn[b] @ Vhn[b]^T via V_WMMA_F32_16X16X4_F32 (K=32 -> 8 wmma)
// grid: (16, 128, 4), block: 256 (8 waves, one 16x16 column tile each)
// ---------------------------------------------------------------------------
__global__ void corr_wmma_kernel(const float* __restrict__ Vn,
                                 const float* __restrict__ Vhn,
                                 float* __restrict__ P) {
    const int lane = threadIdx.x & 31;
    const int w    = threadIdx.x >> 5;
    const int m0   = (blockIdx.x * 8 + w) * 16;   // column tile
    const int n0   = blockIdx.y * 16;             // row tile
    const int b    = blockIdx.z;
    const int rc   = lane & 15;                   // A row / B col within tile
    const int koff = (lane >> 4) * 2;             // K pair per half-wave (ISA 16x4 layout)

    const float* Arow = Vn  + ((size_t)b * Nn + n0 + rc) * KD + koff;
    const float* Brow = Vhn + ((size_t)b * Nn + m0 + rc) * KD + koff;

    v8f acc = {0.f, 0.f, 0.f, 0.f, 0.f, 0.f, 0.f, 0.f};
    #pragma unroll
    for (int kk = 0; kk < 8; ++kk) {
        v2f a  = *(const v2f*)(Arow + kk * 4);
        v2f bb = *(const v2f*)(Brow + kk * 4);
        acc = __builtin_amdgcn_wmma_f32_16x16x4_f32(
            /*neg_a=*/false, a, /*neg_b=*/false, bb,
            /*c_mod=*/(short)0, acc, /*reuse_a=*/false, /*reuse_b=*/false);
    }
    // C/D layout: VGPR r -> row r + 8*(lane>>4), col = lane&15
    size_t base = ((size_t)b * Nn + n0 + 8 * (lane >> 4)) * (size_t)Nn + m0 + rc;
    #pragma unroll
    for (int r = 0; r < 8; ++r) P[base + (size_t)r * Nn] = acc[r];
}

// ---------------------------------------------------------------------------
// K4: pearson_cor_mean = mean over batch (must run before K5 mutates P)
// ---------------------------------------------------------------------------
__global__ void mean_kernel(const float* __restrict__ P, float* __restrict__ meanOut) {
    const size_t NNe = (size_t)Nn * Nn;
    size_t i = (size_t)blockIdx.x * blockDim.x + threadIdx.x;
    if (i < NNe) {
        float s = P[i] + P[i + NNe] + P[i + 2 * NNe] + P[i + 3 * NNe];
        meanOut[i] = 0.25f * s;
    }
}

// ---------------------------------------------------------------------------
// K5: per-row top-32 -> sparse scatter -> softmax, in-place on P
// grid: 8192 blocks (one row), block: 256 threads
// ---------------------------------------------------------------------------
__global__ void topk_softmax_kernel(float* __restrict__ mult) {
    __shared__ float sr[Nn];
    __shared__ float topv[32];
    __shared__ int   topi[32];
    __shared__ float rv[8];
    __shared__ int   ri[8];
    __shared__ float sM, sZ;
    const int t = threadIdx.x;
    float* rowptr = mult + (size_t)blockIdx.x * Nn;
    for (int i = t; i < Nn; i += 256) sr[i] = rowptr[i];
    __syncthreads();

    const float NEG = -__builtin_inff();
    for (int j = 0; j < 32; ++j) {
        // local argmax, tie -> lowest index (ascending scan with strict >)
        float bv = NEG; int bi = 0;
        for (int i = t; i < Nn; i += 256) {
            float v = sr[i];
            if (v > bv) { bv = v; bi = i; }
        }
        // wave butterfly reduce (wave32)
        #pragma unroll
        for (int o = 16; o > 0; o >>= 1) {
            float ov = __shfl_xor(bv, o, 32);
            int   oi = __shfl_xor(bi, o, 32);
            if (ov > bv || (ov == bv && oi < bi)) { bv = ov; bi = oi; }
        }
        if ((t & 31) == 0) { rv[t >> 5] = bv; ri[t >> 5] = bi; }
        __syncthreads();
        if (t == 0) {
            float wv = rv[0]; int wi = ri[0];
            for (int q = 1; q < 8; ++q) {
                if (rv[q] > wv || (rv[q] == wv && ri[q] < wi)) { wv = rv[q]; wi = ri[q]; }
            }
            topv[j] = wv; topi[j] = wi;
            sr[wi] = NEG;                       // exclude from next rounds
        }
        __syncthreads();
    }

    if (t == 0) {
        float M = fmaxf(topv[0], 0.0f);         // sparse row has 2016 zeros
        float Z = (float)(Nn - 32) * expf(-M);
        for (int q = 0; q < 32; ++q) Z += expf(topv[q] - M);
        sM = M; sZ = Z;
    }
    __syncthreads();
    const float invZ = 1.0f / sZ;
    const float base = expf(-sM) * invZ;
    for (int i = t; i < Nn; i += 256) sr[i] = base;
    __syncthreads();
    if (t < 32) sr[topi[t]] = expf(topv[t] - sM) * invZ;
    __syncthreads();
    // vectorized writeback
    float4* dst = (float4*)rowptr;
    const float4* src = (const float4*)sr;
    for (int i = t; i < Nn / 4; i += 256) dst[i] = src[i];
}

// ---------------------------------------------------------------------------
extern "C" void kernel_launch(void* const* d_in, const int* in_sizes, int n_in,
                              void* d_out, int out_size, void* d_ws, size_t ws_size,
                              hipStream_t stream) {
    const float* x      = (const float*)d_in[0];
    const float* target = (const float*)d_in[1];
    const float* conv_w = (const float*)d_in[2];
    const float* proj_w = (const float*)d_in[3];
    const float* proj_b = (const float*)d_in[4];
    const float* W0     = (const float*)d_in[5];
    const float* U0     = (const float*)d_in[6];

    float* out     = (float*)d_out;
    float* emb     = out;                                   // 4*2048*64      = 524288
    float* mult    = out + (size_t)Bn * Nn * DM;            // 4*2048*2048    = 16777216
    float* meanOut = mult + (size_t)Bn * Nn * Nn;           // 2048*2048      = 4194304

    float* Vn  = (float*)d_ws;                              // [4*2048, 32]
    float* Vhn = Vn + (size_t)Bn * Nn * KD;                 // [4*2048, 32]   (2 MB total)

    emb_kernel<<<Bn * Nn, 320, 0, stream>>>(x, conv_w, proj_w, proj_b, emb);
    vec_kernel<<<Bn * Nn, 64, 0, stream>>>(target, W0, U0, Vn, Vhn);
    corr_wmma_kernel<<<dim3(16, 128, Bn), 256, 0, stream>>>(Vn, Vhn, mult);
    mean_kernel<<<(Nn * Nn) / 256, 256, 0, stream>>>(mult, meanOut);
    topk_softmax_kernel<<<Bn * Nn, 256, 0, stream>>>(mult);
}